// Corr2DZNM_70085276336449
// MI455X (gfx1250) — compile-verified
//
#include <hip/hip_runtime.h>
#include <hip/hip_bf16.h>

typedef float v2f __attribute__((ext_vector_type(2)));
typedef float v8f __attribute__((ext_vector_type(8)));
typedef unsigned int v4u __attribute__((ext_vector_type(4)));
typedef int v4i __attribute__((ext_vector_type(4)));
typedef int v8i __attribute__((ext_vector_type(8)));

#define BN   4
#define CN   64
#define HN   256
#define WN   512
#define ND   65
#define MAXD 64
#define PATCH_N 576.0f

// padded dims
#define HP   258            // H + 2*pad
#define WP0  514            // W + 2*pad            (x0 padded width)
#define WP1  642            // W + 2*pad + 2*MAXD   (x1 padded width)
#define WB1  640            // box output width over x1pad

// ---------------------------------------------------------------------------
// Pass 1: per-pixel channel reduction into zero-padded fields P = sum_c v,
//         PQ = sum_c v^2.  xoff = 1 for x0 (pad only), 65 for x1 (pad+MAXD).
// ---------------------------------------------------------------------------
__global__ void reduce_pad_kernel(const float* __restrict__ x,
                                  float* __restrict__ P,
                                  float* __restrict__ PQ,
                                  int Wpad, int xoff, int total) {
  int idx = blockIdx.x * blockDim.x + threadIdx.x;
  if (idx >= total) return;
  int xp  = idx % Wpad;
  int t   = idx / Wpad;
  int yp  = t % HP;
  int b   = t / HP;
  int y   = yp - 1;
  int xg  = xp - xoff;
  float s = 0.f, sq = 0.f;
  if (y >= 0 && y < HN && xg >= 0 && xg < WN) {
    const float* base = x + ((size_t)b * CN * HN + (size_t)y) * WN + xg;
    #pragma unroll 4
    for (int c = 0; c < CN; ++c) {
      float v = base[(size_t)c * HN * WN];
      s += v; sq += v * v;
    }
  }
  P[idx]  = s;
  PQ[idx] = sq;
}

// ---------------------------------------------------------------------------
// Pass 2: 3x3 box of P/PQ -> patch sum S and centered L2 norm L.
// ---------------------------------------------------------------------------
__global__ void box_stats_kernel(const float* __restrict__ P,
                                 const float* __restrict__ PQ,
                                 float* __restrict__ S,
                                 float* __restrict__ L,
                                 int Wpad, int Wout, int total) {
  int idx = blockIdx.x * blockDim.x + threadIdx.x;
  if (idx >= total) return;
  int xo = idx % Wout;
  int t  = idx / Wout;
  int y  = t % HN;
  int b  = t / HN;
  float s = 0.f, sq = 0.f;
  #pragma unroll
  for (int dy = 0; dy < 3; ++dy) {
    const float* pr = P  + ((size_t)b * HP + y + dy) * Wpad + xo;
    const float* qr = PQ + ((size_t)b * HP + y + dy) * Wpad + xo;
    #pragma unroll
    for (int dx = 0; dx < 3; ++dx) { s += pr[dx]; sq += qr[dx]; }
  }
  S[idx] = s;
  L[idx] = sqrtf(fmaxf(sq - s * s * (1.0f / PATCH_N), 0.0f));
}

// ---------------------------------------------------------------------------
// Main fused kernel: banded f32 WMMA GEMM (K = 3 rows x 64 ch) producing
// pd[t][x] for 65 displacements x 130 columns, then 3-wide x-box + ZN
// normalization.  One workgroup = (b, y_out, 128-wide x block).
// 320 threads = 10 waves = (parity 2) x (M-tile 5); 5 banded N-tiles/wave.
// Interior workgroups stage A/B tiles with the Tensor Data Mover
// (tensor_load_to_lds + s_wait_tensorcnt); edge workgroups use a
// bounds-checked scalar staging path (zero padding).
// ---------------------------------------------------------------------------
#define XBLK 128
#define RA   144   // A tile width / LDS row stride (144 % 32 == 16: no conflicts)
#define RB   304   // B tile width / LDS row stride (304 % 32 == 16)

__global__ __launch_bounds__(320) void
corr_wmma_kernel(const float* __restrict__ x0, const float* __restrict__ x1,
                 const float* __restrict__ S0b, const float* __restrict__ L0b,
                 const float* __restrict__ S1b, const float* __restrict__ L1b,
                 float* __restrict__ out) {
  const int xb   = blockIdx.x * XBLK;
  const int y    = blockIdx.y;
  const int b    = blockIdx.z;
  const int tid  = threadIdx.x;
  const int lane = tid & 31;
  const int wv   = tid >> 5;      // 0..9
  const int par  = wv & 1;        // column parity
  const int mt   = wv >> 1;       // M tile 0..4

  __shared__ float As[CN * RA + 32];   // x0p tile, row-major [c][xpl]
  __shared__ float Bs[CN * RB];        // x1pad tile, row-major [c][jl]
  __shared__ float pd[ND * 132];       // band dot sums [t][xpl]
  __shared__ float s0row[128], l0row[128];
  __shared__ float s1row[256], l1row[256];

  // stage normalization stat rows
  for (int i = tid; i < 128; i += 320) {
    s0row[i] = S0b[((size_t)b * HN + y) * WN + xb + i];
    l0row[i] = L0b[((size_t)b * HN + y) * WN + xb + i];
  }
  for (int i = tid; i < 256; i += 320) {
    s1row[i] = S1b[((size_t)b * HN + y) * WB1 + xb + i];
    l1row[i] = L1b[((size_t)b * HN + y) * WB1 + xb + i];
  }

  v8f acc[5];
  v8f zero = {};
  #pragma unroll
  for (int q = 0; q < 5; ++q) acc[q] = zero;

  const int half  = lane >> 4;
  const int r     = lane & 15;
  const int xpl_a = par + 2 * (16 * mt + r);     // A fragment column

  // TDM is usable when every global address of the padded tile is in-bounds:
  // A needs [xb-1, xb+143) in [0,512), B needs [xb-65, xb+239) in [0,512),
  // and all three rows y-1..y+1 valid.  Uniform per-block condition.
  const bool tdm_ok = (xb >= MAXD + 1) && (xb + RB - (MAXD + 1) <= WN) &&
                      (y >= 1) && (y <= HN - 2);

  for (int dy = 0; dy < 3; ++dy) {
    const int yr = y + dy - 1;
    if (tdm_ok) {
      if (wv == 0) {
        const v4i gz4 = {0, 0, 0, 0};
        const v8i gz8 = {0, 0, 0, 0, 0, 0, 0, 0};
        // ---- A tile: 144 cols x 64 channels ----
        {
          unsigned ldsA = (unsigned)(size_t)&As[0];
          unsigned long long ga = (unsigned long long)(size_t)
              (x0 + (((size_t)b * CN) * HN + yr) * WN + (xb - 1));
          v4u g0 = { 1u,                                   // count=1 (valid)
                     ldsA,                                 // lds_addr
                     (unsigned)ga,                         // global_addr[31:0]
                     ((unsigned)(ga >> 32) & 0x01FFFFFFu) | 0x80000000u }; // [56:32] | type=2
          v8i g1 = { (int)(2u << 16),                      // data_size=4B
                     (int)((unsigned)RA << 16),            // tensor_dim0 = RA
                     (int)(64u << 16),                     // tensor_dim1 = 64
                     (int)((unsigned)RA << 16),            // tile_dim0 = RA
                     64,                                   // tile_dim1 = 64
                     (int)(HN * WN),                       // tensor_dim0_stride
                     0, 0 };
          __builtin_amdgcn_tensor_load_to_lds(g0, g1, gz4, gz4, gz8, 0);
        }
        // ---- B tile: 304 cols x 64 channels ----
        {
          unsigned ldsB = (unsigned)(size_t)&Bs[0];
          unsigned long long ga = (unsigned long long)(size_t)
              (x1 + (((size_t)b * CN) * HN + yr) * WN + (xb - (MAXD + 1)));
          v4u g0 = { 1u, ldsB, (unsigned)ga,
                     ((unsigned)(ga >> 32) & 0x01FFFFFFu) | 0x80000000u };
          v8i g1 = { (int)(2u << 16),
                     (int)((unsigned)RB << 16),
                     (int)(64u << 16),
                     (int)((unsigned)RB << 16),
                     64,
                     (int)(HN * WN),
                     0, 0 };
          __builtin_amdgcn_tensor_load_to_lds(g0, g1, gz4, gz4, gz8, 0);
        }
        __builtin_amdgcn_s_wait_tensorcnt(0);
      }
      __syncthreads();
    } else {
      const bool rowok = (yr >= 0) && (yr < HN);
      // stage A tile: 130 used columns x 64 channels (8320 = 26*320 exactly)
      for (int i = tid; i < 130 * CN; i += 320) {
        int xpl = i % 130, c = i / 130;
        int xg = xb + xpl - 1;
        float v = 0.f;
        if (rowok && xg >= 0 && xg < WN)
          v = x0[(((size_t)b * CN + c) * HN + yr) * WN + xg];
        As[c * RA + xpl] = v;
      }
      // stage B tile: 288 used columns x 64 channels
      for (int i = tid; i < 288 * CN; i += 320) {
        int jl = i % 288, c = i / 288;
        int xg = xb + jl - (MAXD + 1);
        float v = 0.f;
        if (rowok && xg >= 0 && xg < WN)
          v = x1[(((size_t)b * CN + c) * HN + yr) * WN + xg];
        Bs[c * RB + jl] = v;
      }
      __syncthreads();
    }

    // banded GEMM for this K-chunk: 16 steps of K=4
    #pragma unroll 4
    for (int kc = 0; kc < CN; kc += 4) {
      const int k0 = kc + 2 * half;
      v2f a = { As[k0 * RA + xpl_a], As[(k0 + 1) * RA + xpl_a] };
      #pragma unroll
      for (int q = 0; q < 5; ++q) {
        const int nt = mt + q;
        const int jl = par + 2 * (16 * nt + r);
        v2f bb = { Bs[k0 * RB + jl], Bs[(k0 + 1) * RB + jl] };
        acc[q] = __builtin_amdgcn_wmma_f32_16x16x4_f32(
            false, a, false, bb, (short)0, acc[q], false, false);
      }
    }
    __syncthreads();
  }

  // scatter accumulators into the displacement-indexed band buffer
  #pragma unroll
  for (int q = 0; q < 5; ++q) {
    #pragma unroll
    for (int v = 0; v < 8; ++v) {
      int m   = v + 8 * half;            // row within tile
      int mg  = 16 * mt + m;
      int xpl = par + 2 * mg;
      int t   = 16 * q + r - m;          // displacement index
      if (t >= 0 && t < ND && xpl < 130)
        pd[t * 132 + xpl] = acc[q][v];
    }
  }
  __syncthreads();

  // 3-wide x box + zero-normalization + store (8320 = 26*320 exactly)
  const float inv_n = 1.0f / PATCH_N;
  for (int i = tid; i < ND * XBLK; i += 320) {
    int t = i >> 7, xo = i & 127;
    float dot = pd[t * 132 + xo] + pd[t * 132 + xo + 1] + pd[t * 132 + xo + 2];
    float s0 = s0row[xo], l0 = l0row[xo];
    float s1 = s1row[xo + 2 * t], l1 = l1row[xo + 2 * t];
    float val = (dot - s0 * s1 * inv_n) / (l0 * l1 + 1e-8f);
    out[(((size_t)b * ND + t) * HN + y) * WN + xb + xo] = val;
  }
}

// ---------------------------------------------------------------------------
extern "C" void kernel_launch(void* const* d_in, const int* in_sizes, int n_in,
                              void* d_out, int out_size, void* d_ws, size_t ws_size,
                              hipStream_t stream) {
  const float* x0 = (const float*)d_in[0];
  const float* x1 = (const float*)d_in[1];
  float* out = (float*)d_out;

  // workspace layout (floats)
  float* ws  = (float*)d_ws;
  const size_t n_p0 = (size_t)BN * HP * WP0;   // 530448
  const size_t n_p1 = (size_t)BN * HP * WP1;   // 662544
  const size_t n_s0 = (size_t)BN * HN * WN;    // 524288
  const size_t n_s1 = (size_t)BN * HN * WB1;   // 655360
  float* P0  = ws;
  float* Q0  = P0 + n_p0;
  float* P1  = Q0 + n_p0;
  float* Q1  = P1 + n_p1;
  float* S0b = Q1 + n_p1;
  float* L0b = S0b + n_s0;
  float* S1b = L0b + n_s0;
  float* L1b = S1b + n_s1;

  // Pass 1: channel reductions into padded fields
  {
    int total = (int)n_p0;
    reduce_pad_kernel<<<(total + 255) / 256, 256, 0, stream>>>(
        x0, P0, Q0, WP0, 1, total);
  }
  {
    int total = (int)n_p1;
    reduce_pad_kernel<<<(total + 255) / 256, 256, 0, stream>>>(
        x1, P1, Q1, WP1, MAXD + 1, total);
  }
  // Pass 2: 3x3 box stats (patch sum + centered norm)
  {
    int total = (int)n_s0;
    box_stats_kernel<<<(total + 255) / 256, 256, 0, stream>>>(
        P0, Q0, S0b, L0b, WP0, WN, total);
  }
  {
    int total = (int)n_s1;
    box_stats_kernel<<<(total + 255) / 256, 256, 0, stream>>>(
        P1, Q1, S1b, L1b, WP1, WB1, total);
  }
  // Main fused WMMA correlation
  dim3 grid(WN / XBLK, HN, BN);   // (4, 256, 4)
  corr_wmma_kernel<<<grid, 320, 0, stream>>>(x0, x1, S0b, L0b, S1b, L1b, out);
}